// FRMIL_11759620456986
// MI455X (gfx1250) — compile-verified
//
#include <hip/hip_runtime.h>
#include <math.h>

typedef float v2f __attribute__((ext_vector_type(2)));
typedef float v4f __attribute__((ext_vector_type(4)));
typedef float v8f __attribute__((ext_vector_type(8)));
typedef int   vsi4 __attribute__((vector_size(4 * sizeof(int))));   // matches builtin param

static constexpr int    N_ROWS  = 60000;
static constexpr int    D       = 512;
static constexpr int    H_GRID  = 245;               // ceil(sqrt(N)) ; 245*245 = 60025
static constexpr int    SEQ     = 60026;             // cls + 60025 spatial positions
static constexpr int    SEQ_PAD = 60032;             // 16 * 3752 (WMMA tiling)
static constexpr int    NTILE   = SEQ_PAD / 16;      // 3752
static constexpr int    NHEADS  = 8;
static constexpr int    NB_WS   = (SEQ + 255) / 256; // 235 weighted-sum blocks
static constexpr float  SCALE   = 0.04419417382415922f; // 1/sqrt(512)

// ---- workspace layout (float offsets) ----
static constexpr size_t WS_XSEQ   = 0;
static constexpr size_t WS_SCORES = WS_XSEQ   + (size_t)SEQ_PAD * D;   // [SEQ_PAD][16]
static constexpr size_t WS_RMAT   = WS_SCORES + (size_t)SEQ_PAD * 16;  // packed [512][16]
static constexpr size_t WS_QIN    = WS_RMAT   + 512 * 16;              // gathered inputs[idx]
static constexpr size_t WS_QPROJ  = WS_QIN    + 512;                   // q = Q@w_q + b_q
static constexpr size_t WS_CVEC   = WS_QPROJ  + 512;                   // per-head score bias
static constexpr size_t WS_HMAX   = WS_CVEC   + 16;
static constexpr size_t WS_HSUM   = WS_HMAX   + 16;
static constexpr size_t WS_UPART  = WS_HSUM   + 16;                    // [NB_WS][8][512]
static constexpr size_t WS_UMAT   = WS_UPART  + (size_t)NB_WS * NHEADS * D;
static constexpr size_t WS_SLOT   = WS_UMAT   + NHEADS * D;            // u64 argmax slot

#if defined(__has_builtin)
#if __has_builtin(__builtin_amdgcn_global_load_async_to_lds_b128) && \
    __has_builtin(__builtin_amdgcn_s_wait_asynccnt)
#define USE_ASYNC_LDS 1
#endif
#endif

// ---------------- K0: per-call init ----------------
__global__ void k0_init(float* __restrict__ ws) {
    if (blockIdx.x == 0 && threadIdx.x == 0)
        *(unsigned long long*)(ws + WS_SLOT) = 0ull;
    int idx = blockIdx.x * 256 + threadIdx.x;       // zero xseq pad rows
    if (idx < (SEQ_PAD - SEQ) * D)
        ws[WS_XSEQ + (size_t)SEQ * D + idx] = 0.f;
}

// ---------------- K1: encoder GEMV + sigmoid + packed argmax ----------------
__global__ void k1_enc(const float* __restrict__ inputs, const float* __restrict__ w_enc,
                       const float* __restrict__ b_enc, float* __restrict__ a1_out,
                       unsigned long long* __restrict__ slot) {
    int wave = threadIdx.x >> 5, lane = threadIdx.x & 31;
    int row  = blockIdx.x * 8 + wave;               // grid 7500 * 8 waves = 60000 rows
    const float* xr = inputs + (size_t)row * D + lane * 16;
    const float* wr = w_enc + lane * 16;
    float s = 0.f;
#pragma unroll
    for (int i = 0; i < 16; i += 4) {
        v4f a = *(const v4f*)(xr + i);
        v4f w = *(const v4f*)(wr + i);
        s = fmaf(a.x, w.x, fmaf(a.y, w.y, fmaf(a.z, w.z, fmaf(a.w, w.w, s))));
    }
    for (int off = 16; off; off >>= 1) s += __shfl_down(s, off, 32);
    if (lane == 0) {
        float act = 1.f / (1.f + __expf(-(s + b_enc[0])));
        a1_out[row] = act;
        unsigned long long pack =
            ((unsigned long long)__float_as_uint(act) << 32) |
            (unsigned)(0xFFFFFFFFu - (unsigned)row);         // ties -> smallest row
        atomicMax(slot, pack);
    }
}

// ---------------- K2: decode argmax, gather Q row ----------------
__global__ void k2_pick(const float* __restrict__ inputs,
                        const float* __restrict__ ws_ro, float* __restrict__ qin) {
    __shared__ unsigned sidx;
    if (threadIdx.x == 0) {
        unsigned long long pack = *(const unsigned long long*)(ws_ro + WS_SLOT);
        sidx = 0xFFFFFFFFu - (unsigned)(pack & 0xFFFFFFFFull);
    }
    __syncthreads();
    const float* src = inputs + (size_t)sidx * D;
    for (int c = threadIdx.x; c < D; c += blockDim.x) qin[c] = src[c];
}

// ---------------- K3: i_shift = relu(inputs - Q) ----------------
__global__ void k3_shift(const float* __restrict__ inputs, const float* __restrict__ qin,
                         float* __restrict__ ishift) {
    int row = blockIdx.x, c = threadIdx.x * 2;
    if (threadIdx.x < 16 && row + 64 < N_ROWS)       // warm lines 64 rows ahead
        __builtin_prefetch(inputs + (size_t)(row + 64) * D + threadIdx.x * 32, 0, 0);
    v2f x = *(const v2f*)(inputs + (size_t)row * D + c);
    v2f q = *(const v2f*)(qin + c);
    v2f r; r.x = fmaxf(x.x - q.x, 0.f); r.y = fmaxf(x.y - q.y, 0.f);
    *(v2f*)(ishift + (size_t)row * D + c) = r;
}

// ---------------- K4: depthwise 3x3 conv + residual -> xseq rows; cls -> row 0 ----------------
__global__ void k4_conv(const float* __restrict__ ishift, const float* __restrict__ conv_w,
                        const float* __restrict__ conv_b, const float* __restrict__ cls_tok,
                        float* __restrict__ xseq) {
    int c = threadIdx.x * 2;
    if (blockIdx.x == 0) {                           // cls token row
        *(v2f*)(xseq + c) = *(const v2f*)(cls_tok + c);
        return;
    }
    int p = blockIdx.x - 1;                          // spatial position 0..60024
    int y = p / H_GRID, x = p - y * H_GRID;
    float w0[9], w1[9];
#pragma unroll
    for (int k = 0; k < 9; ++k) { w0[k] = conv_w[(size_t)c * 9 + k];
                                  w1[k] = conv_w[(size_t)(c + 1) * 9 + k]; }
    v2f bb = *(const v2f*)(conv_b + c);
    float ax = bb.x, ay = bb.y, cx = 0.f, cy = 0.f;
#pragma unroll
    for (int ky = 0; ky < 3; ++ky)
#pragma unroll
        for (int kx = 0; kx < 3; ++kx) {
            int yy = y + ky - 1, xx = x + kx - 1;
            float tx = 0.f, ty = 0.f;
            if (yy >= 0 && yy < H_GRID && xx >= 0 && xx < H_GRID) {
                int pp = yy * H_GRID + xx;
                if (pp >= N_ROWS) pp -= N_ROWS;      // xp wraps first 25 rows
                v2f t = *(const v2f*)(ishift + (size_t)pp * D + c);
                tx = t.x; ty = t.y;
            }
            if (ky == 1 && kx == 1) { cx = tx; cy = ty; }
            ax = fmaf(w0[ky * 3 + kx], tx, ax);
            ay = fmaf(w1[ky * 3 + kx], ty, ay);
        }
    v2f acc; acc.x = ax + cx; acc.y = ay + cy;       // residual
    *(v2f*)(xseq + (size_t)blockIdx.x * D + c) = acc;
}

// ---------------- K5: q = Q@w_q+b_q ; R (packed WMMA-B layout) ; c_h bias ----------------
__global__ void k5_qr(const float* __restrict__ qin, const float* __restrict__ w_q,
                      const float* __restrict__ b_q, const float* __restrict__ w_k,
                      const float* __restrict__ b_k, float* __restrict__ qproj,
                      float* __restrict__ rmat, float* __restrict__ cvec) {
    __shared__ float qis[D];
    __shared__ float qs[D];
    int t = threadIdx.x;
    qis[t] = qin[t]; qis[t + 256] = qin[t + 256];
    __syncthreads();
#pragma unroll
    for (int rep = 0; rep < 2; ++rep) {
        int col = t + rep * 256;
        float s = b_q[col];
        for (int d = 0; d < D; ++d) s = fmaf(qis[d], w_q[(size_t)d * D + col], s);
        qs[col] = s; qproj[col] = s;
    }
    __syncthreads();
    for (int e = t; e < 512 * 16; e += 256) {        // R[i][h], head-padded to 16
        int i = e >> 4, h = e & 15;
        float r = 0.f;
        if (h < NHEADS) {
            float s = 0.f;
            const float* wr = w_k + (size_t)i * D + h * 64;
            const float* qh = qs + h * 64;
            for (int d = 0; d < 64; ++d) s = fmaf(wr[d], qh[d], s);
            r = s * SCALE;
        }
        // write directly in WMMA-B packed layout: [(i>>1)*32 + h*2 + (i&1)]
        rmat[(i >> 1) * 32 + h * 2 + (i & 1)] = r;
    }
    if (t < 16) {
        float cval = 0.f;
        if (t < NHEADS) {
            float s = 0.f;
            for (int d = 0; d < 64; ++d) s = fmaf(b_k[t * 64 + d], qs[t * 64 + d], s);
            cval = s * SCALE;
        }
        cvec[t] = cval;
    }
}

// ---------------- K6: scores = xseq @ R  via V_WMMA_F32_16X16X4_F32 ----------------
__global__ void k6_scores(const float* __restrict__ xseq, const float* __restrict__ rmat,
                          const float* __restrict__ cvec, float* __restrict__ scores) {
    __shared__ __align__(16) float Rs[512 * 16];     // already packed in ws by K5
    int t = threadIdx.x;
#ifdef USE_ASYNC_LDS
    {   // verbatim async copy: 8192 floats = 256 thr * 8 * b128
        auto* gsrc = (__attribute__((address_space(1))) vsi4*)(rmat + (size_t)t * 32);
        auto* ldst = (__attribute__((address_space(3))) vsi4*)(&Rs[t * 32]);
        __builtin_amdgcn_global_load_async_to_lds_b128(gsrc, ldst, 0,   0);
        __builtin_amdgcn_global_load_async_to_lds_b128(gsrc, ldst, 16,  0);
        __builtin_amdgcn_global_load_async_to_lds_b128(gsrc, ldst, 32,  0);
        __builtin_amdgcn_global_load_async_to_lds_b128(gsrc, ldst, 48,  0);
        __builtin_amdgcn_global_load_async_to_lds_b128(gsrc, ldst, 64,  0);
        __builtin_amdgcn_global_load_async_to_lds_b128(gsrc, ldst, 80,  0);
        __builtin_amdgcn_global_load_async_to_lds_b128(gsrc, ldst, 96,  0);
        __builtin_amdgcn_global_load_async_to_lds_b128(gsrc, ldst, 112, 0);
        __builtin_amdgcn_s_wait_asynccnt(0);
    }
#else
    for (int e = t; e < 512 * 16; e += 256) Rs[e] = rmat[e];
#endif
    __syncthreads();
    int wave = t >> 5, lane = t & 31;
    int m = lane & 15, hi = lane >> 4;               // n == m for B/C fragments
    int row0 = (blockIdx.x * 8 + wave) * 16;
    const float* xrow = xseq + (size_t)(row0 + m) * D + 2 * hi;
    v8f c = {0.f, 0.f, 0.f, 0.f, 0.f, 0.f, 0.f, 0.f};
#pragma unroll 8
    for (int k = 0; k < 128; ++k) {                  // K = 512 = 128 * 4
        v2f a = *(const v2f*)(xrow + 4 * k);         // A 16x4 f32: V0=K0/K2, V1=K1/K3
        v2f b = *(const v2f*)(&Rs[(2 * k + hi) * 32 + 2 * m]);
        c = __builtin_amdgcn_wmma_f32_16x16x4_f32(false, a, false, b,
                                                  (short)0, c, false, false);
    }
    float cv = cvec[m];
#pragma unroll
    for (int i = 0; i < 8; ++i)                      // C: VGPR i -> rows i / i+8
        scores[(size_t)(row0 + i + 8 * hi) * 16 + m] = c[i] + cv;
}

// ---------------- K7: per-head softmax max & sum ----------------
__global__ void k7_stats(const float* __restrict__ scores, float* __restrict__ hmax,
                         float* __restrict__ hsum) {
    int h = blockIdx.x, t = threadIdx.x;
    __shared__ float red[256];
    float m = -3.4e38f;
    for (int j = t; j < SEQ; j += 256) m = fmaxf(m, scores[(size_t)j * 16 + h]);
    red[t] = m; __syncthreads();
    for (int s = 128; s; s >>= 1) { if (t < s) red[t] = fmaxf(red[t], red[t + s]); __syncthreads(); }
    float mx = red[0]; __syncthreads();
    float sum = 0.f;
    for (int j = t; j < SEQ; j += 256) sum += __expf(scores[(size_t)j * 16 + h] - mx);
    red[t] = sum; __syncthreads();
    for (int s = 128; s; s >>= 1) { if (t < s) red[t] += red[t + s]; __syncthreads(); }
    if (t == 0) { hmax[h] = mx; hsum[h] = red[0]; }
}

// ---------------- K8: u_part = sum_j exp(s-max) * xseq_j (per block partials) ----------------
__global__ void k8_wsum(const float* __restrict__ xseq, const float* __restrict__ scores,
                        const float* __restrict__ hmax, float* __restrict__ upart) {
    __shared__ float sE[256][NHEADS];
    int t = threadIdx.x, j0 = blockIdx.x * 256, j = j0 + t;
    if (j < SEQ) {
#pragma unroll
        for (int h = 0; h < NHEADS; ++h)
            sE[t][h] = __expf(scores[(size_t)j * 16 + h] - hmax[h]);
    } else {
#pragma unroll
        for (int h = 0; h < NHEADS; ++h) sE[t][h] = 0.f;
    }
    __syncthreads();
    int cnt = min(256, SEQ - j0);
    int c = t * 2;
    float accx[NHEADS] = {0}, accy[NHEADS] = {0};
    for (int jj = 0; jj < cnt; ++jj) {
        v2f xv = *(const v2f*)(xseq + (size_t)(j0 + jj) * D + c);
#pragma unroll
        for (int h = 0; h < NHEADS; ++h) {
            float e = sE[jj][h];                     // LDS broadcast
            accx[h] = fmaf(e, xv.x, accx[h]);
            accy[h] = fmaf(e, xv.y, accy[h]);
        }
    }
    float* up = upart + (size_t)blockIdx.x * NHEADS * D;
#pragma unroll
    for (int h = 0; h < NHEADS; ++h) { up[h * D + c] = accx[h]; up[h * D + c + 1] = accy[h]; }
}

// ---------------- K8b: reduce partials (fixed order -> deterministic) ----------------
__global__ void k8b_ured(const float* __restrict__ upart, float* __restrict__ umat) {
    int idx = blockIdx.x * 256 + threadIdx.x;        // 4096 outputs
    float s = 0.f;
    for (int b = 0; b < NB_WS; ++b) s += upart[(size_t)b * NHEADS * D + idx];
    umat[idx] = s;
}

// ---------------- K9: O = q + (u/sum)@w_v + b_v ; O += relu(O@w_o+b_o) ; logits ----------------
__global__ void k9_final(const float* __restrict__ qproj, const float* __restrict__ umat,
                         const float* __restrict__ hsum, const float* __restrict__ w_v,
                         const float* __restrict__ b_v, const float* __restrict__ w_o,
                         const float* __restrict__ b_o, const float* __restrict__ w_fc,
                         const float* __restrict__ b_fc, float* __restrict__ out) {
    __shared__ float Us[NHEADS * D];
    __shared__ float inv[NHEADS];
    __shared__ float Os[D];
    __shared__ float O2s[D];
    int t = threadIdx.x;
    for (int e = t; e < NHEADS * D; e += 256) Us[e] = umat[e];
    if (t < NHEADS) inv[t] = 1.f / hsum[t];
    __syncthreads();
#pragma unroll
    for (int rep = 0; rep < 2; ++rep) {
        int col = t + rep * 256;
        int h = col >> 6;
        float s = 0.f;
        const float* uh = Us + h * D;
        for (int i = 0; i < D; ++i) s = fmaf(uh[i], w_v[(size_t)i * D + col], s);
        Os[col] = qproj[col] + b_v[col] + s * inv[h];
    }
    __syncthreads();
#pragma unroll
    for (int rep = 0; rep < 2; ++rep) {
        int col = t + rep * 256;
        float s = b_o[col];
        for (int i = 0; i < D; ++i) s = fmaf(Os[i], w_o[(size_t)i * D + col], s);
        O2s[col] = Os[col] + fmaxf(s, 0.f);
    }
    __syncthreads();
    if (t < 2) {
        float s = b_fc[t];
        for (int i = 0; i < D; ++i) s = fmaf(O2s[i], w_fc[(size_t)i * 2 + t], s);
        out[t] = s;
    }
}

extern "C" void kernel_launch(void* const* d_in, const int* in_sizes, int n_in,
                              void* d_out, int out_size, void* d_ws, size_t ws_size,
                              hipStream_t stream) {
    (void)in_sizes; (void)n_in; (void)out_size; (void)ws_size;
    const float* inputs  = (const float*)d_in[0];
    const float* w_enc   = (const float*)d_in[1];
    const float* b_enc   = (const float*)d_in[2];
    const float* cls_tok = (const float*)d_in[3];
    const float* conv_w  = (const float*)d_in[4];
    const float* conv_b  = (const float*)d_in[5];
    const float* w_q     = (const float*)d_in[6];
    const float* b_q     = (const float*)d_in[7];
    const float* w_k     = (const float*)d_in[8];
    const float* b_k     = (const float*)d_in[9];
    const float* w_v     = (const float*)d_in[10];
    const float* b_v     = (const float*)d_in[11];
    const float* w_o     = (const float*)d_in[12];
    const float* b_o     = (const float*)d_in[13];
    const float* w_fc    = (const float*)d_in[14];
    const float* b_fc    = (const float*)d_in[15];

    float* out    = (float*)d_out;
    float* ishift = out + 2;                              // (1,60000,512)
    float* a1     = out + 2 + (size_t)N_ROWS * D;         // (1,60000)

    float* ws     = (float*)d_ws;
    float* xseq   = ws + WS_XSEQ;
    float* scores = ws + WS_SCORES;
    float* rmat   = ws + WS_RMAT;
    float* qin    = ws + WS_QIN;
    float* qproj  = ws + WS_QPROJ;
    float* cvec   = ws + WS_CVEC;
    float* hmax   = ws + WS_HMAX;
    float* hsum   = ws + WS_HSUM;
    float* upart  = ws + WS_UPART;
    float* umat   = ws + WS_UMAT;
    unsigned long long* slot = (unsigned long long*)(ws + WS_SLOT);

    k0_init <<<12, 256, 0, stream>>>(ws);
    k1_enc  <<<N_ROWS / 8, 256, 0, stream>>>(inputs, w_enc, b_enc, a1, slot);
    k2_pick <<<1, 256, 0, stream>>>(inputs, ws, qin);
    k3_shift<<<N_ROWS, 256, 0, stream>>>(inputs, qin, ishift);
    k4_conv <<<SEQ, 256, 0, stream>>>(ishift, conv_w, conv_b, cls_tok, xseq);
    k5_qr   <<<1, 256, 0, stream>>>(qin, w_q, b_q, w_k, b_k, qproj, rmat, cvec);
    k6_scores<<<NTILE / 8, 256, 0, stream>>>(xseq, rmat, cvec, scores);
    k7_stats<<<NHEADS, 256, 0, stream>>>(scores, hmax, hsum);
    k8_wsum <<<NB_WS, 256, 0, stream>>>(xseq, scores, hmax, upart);
    k8b_ured<<<NHEADS * D / 256, 256, 0, stream>>>(upart, umat);
    k9_final<<<1, 256, 0, stream>>>(qproj, umat, hsum, w_v, b_v, w_o, b_o, w_fc, b_fc, out);
}